// MambaTritonSSM_85942295593455
// MI455X (gfx1250) — compile-verified
//
#include <hip/hip_runtime.h>
#include <math.h>

// ---------------- problem constants ----------------
#define Bq   4
#define Lq   2048
#define Dq   1024
#define Sq   64
#define LCq  2051                 // L + 3 (conv output length)
#define Tq   128                  // scan chunk length
#define NCq  17                   // ceil(LC / T)
#define LCPq 2176                 // NC * T, padded sequence length
#define XPAD 1028                 // LDS row stride (floats) for 16x1024 tile

typedef __attribute__((ext_vector_type(2))) float v2f;
typedef __attribute__((ext_vector_type(8))) float v8f;

// D = A(16x4 f32) x B(4x16 f32) + C(16x16 f32)   -> v_wmma_f32_16x16x4_f32
__device__ __forceinline__ v8f wmma4(v2f a, v2f b, v8f c) {
  return __builtin_amdgcn_wmma_f32_16x16x4_f32(false, a, false, b, (short)0, c,
                                               false, false);
}

__device__ __forceinline__ float silu_f(float x) {
  return x / (1.0f + expf(-x));
}

// ============================================================
// K1: depthwise conv1d, kernel 4, pad 3 -> xc [B, LCP, D] (pad rows zeroed)
// ============================================================
__global__ __launch_bounds__(256) void k_conv(const float* __restrict__ x,
                                              const float* __restrict__ cw,
                                              const float* __restrict__ cb,
                                              float* __restrict__ xc) {
  size_t idx = (size_t)blockIdx.x * 256 + threadIdx.x;
  if (idx >= (size_t)Bq * LCPq * Dq) return;
  int d = (int)(idx & (Dq - 1));
  size_t r = idx >> 10;                 // row = b*LCP + t
  int t = (int)(r % LCPq);
  int b = (int)(r / LCPq);
  float acc = 0.0f;
  if (t < LCq) {
    acc = cb[d];
#pragma unroll
    for (int k = 0; k < 4; ++k) {
      int ti = t - 3 + k;
      if (ti >= 0 && ti < Lq)
        acc += x[((size_t)b * Lq + ti) * Dq + d] * cw[d * 4 + k];
    }
  }
  xc[idx] = acc;
}

// ============================================================
// K2: WMMA projections. rows = B*LCP (=8704, /16 = 544 tiles)
//   Bc = (xc@W_B^T + b_B) * (xc@W_U^T) ; Cc = xc@W_C^T + b_C   [rows, S]
// block = 128 thr (4 waves); wave w owns S-cols [16w,16w+16); LDS-staged A tile
// ============================================================
__global__ __launch_bounds__(128) void k_proj(const float* __restrict__ xc,
                                              const float* __restrict__ W_U,
                                              const float* __restrict__ W_B,
                                              const float* __restrict__ b_B,
                                              const float* __restrict__ W_C,
                                              const float* __restrict__ b_C,
                                              float* __restrict__ Bc,
                                              float* __restrict__ Cc) {
  __shared__ float tile[16 * XPAD];     // 16 x 1024 activations, padded stride
  const int row0 = blockIdx.x * 16;
  const int tid  = threadIdx.x;
  const int lane = tid & 31;
  const int wave = tid >> 5;

  // cooperative stage: 16 rows x 1024 f32 (4096 float4's)
  for (int i = tid; i < 16 * 256; i += 128) {
    int r  = i >> 8;
    int c4 = i & 255;
    float4 v = ((const float4*)(xc + (size_t)(row0 + r) * Dq))[c4];
    *((float4*)&tile[r * XPAD + c4 * 4]) = v;
  }
  __syncthreads();

  const int n0    = wave * 16;
  const int mrow  = lane & 15;
  const int khalf = (lane >> 4) * 2;    // 0 or 2
  v8f accU = {}, accB = {}, accC = {};

  for (int k0 = 0; k0 < Dq; k0 += 4) {
    const int k = k0 + khalf;
    v2f a;
    a.x = tile[mrow * XPAD + k];
    a.y = tile[mrow * XPAD + k + 1];
    const size_t wo = (size_t)(n0 + mrow) * Dq + k;
    v2f bu; bu.x = W_U[wo]; bu.y = W_U[wo + 1];
    v2f bb; bb.x = W_B[wo]; bb.y = W_B[wo + 1];
    v2f bc; bc.x = W_C[wo]; bc.y = W_C[wo + 1];
    accU = wmma4(a, bu, accU);
    accB = wmma4(a, bb, accB);
    accC = wmma4(a, bc, accC);
  }

  const int ncol = n0 + mrow;
  const float biasB = b_B[ncol];
  const float biasC = b_C[ncol];
#pragma unroll
  for (int v = 0; v < 8; ++v) {
    const int r = row0 + v + ((lane >> 4) << 3);
    Bc[(size_t)r * Sq + ncol] = (accB[v] + biasB) * accU[v];
    Cc[(size_t)r * Sq + ncol] = accC[v] + biasC;
  }
}

// ============================================================
// K3: delta path. one wave per padded row -> LG[row, s] = delta * (-exp(a_hat[s]))
// ============================================================
__global__ __launch_bounds__(256) void k_delta(const float* __restrict__ xc,
                                               const float* __restrict__ a_hat,
                                               const float* __restrict__ W_d1,
                                               const float* __restrict__ b_d1,
                                               const float* __restrict__ W_d2,
                                               const float* __restrict__ b_d2,
                                               float* __restrict__ LG) {
  const int row  = blockIdx.x * 8 + (threadIdx.x >> 5);
  const int lane = threadIdx.x & 31;
  const int t    = row % LCPq;
  const float* xr = xc + (size_t)row * Dq;

  float p0 = 0.f, p1 = 0.f, p2 = 0.f, p3 = 0.f;
  for (int k = lane; k < Dq; k += 32) {
    const float xv = xr[k];
    p0 += xv * W_d1[k];
    p1 += xv * W_d1[Dq + k];
    p2 += xv * W_d1[2 * Dq + k];
    p3 += xv * W_d1[3 * Dq + k];
  }
#pragma unroll
  for (int off = 16; off > 0; off >>= 1) {
    p0 += __shfl_xor(p0, off, 32);
    p1 += __shfl_xor(p1, off, 32);
    p2 += __shfl_xor(p2, off, 32);
    p3 += __shfl_xor(p3, off, 32);
  }
  const float z = silu_f(p0 + b_d1[0]) * W_d2[0] +
                  silu_f(p1 + b_d1[1]) * W_d2[1] +
                  silu_f(p2 + b_d1[2]) * W_d2[2] +
                  silu_f(p3 + b_d1[3]) * W_d2[3] + b_d2[0];
  const float delta = (z > 20.0f) ? z : log1pf(expf(z));  // softplus
  const bool valid = (t < LCq);
  const float v0 = valid ? delta * (-expf(a_hat[lane]))      : 0.0f;
  const float v1 = valid ? delta * (-expf(a_hat[lane + 32])) : 0.0f;
  LG[(size_t)row * Sq + lane]      = v0;
  LG[(size_t)row * Sq + lane + 32] = v1;
}

// ============================================================
// K4: per-chunk inclusive cumsum of logA.  grid = B*NC, block = 64 (thread = s)
// ============================================================
__global__ __launch_bounds__(64) void k_cumsum(const float* __restrict__ LG,
                                               float* __restrict__ CUM) {
  const int bc = blockIdx.x;
  const int s  = threadIdx.x;
  const int b = bc / NCq, c = bc % NCq;
  const float* src = LG + ((size_t)b * LCPq + c * Tq) * Sq + s;
  float* dst = CUM + (size_t)bc * Tq * Sq + s;
  float acc = 0.0f;
  for (int t = 0; t < Tq; ++t) {
    acc += src[(size_t)t * Sq];
    dst[(size_t)t * Sq] = acc;
  }
}

// ============================================================
// K5: segsum attention matrix + Ccum. grid = B*NC*T (one t-row), block = 128 (tau)
//   M[t,tau] = sum_s C_t[s] B_tau[s] exp(cum_t[s]-cum_tau[s])  (tau<=t, exps<=0)
//   Ccum[t,s] = C_t[s] * exp(cum_t[s])
// ============================================================
__global__ __launch_bounds__(128) void k_segsum(const float* __restrict__ Bc,
                                                const float* __restrict__ Cc,
                                                const float* __restrict__ CUM,
                                                float* __restrict__ Mmat,
                                                float* __restrict__ Ccum) {
  __shared__ float sC[Sq], sCt[Sq];
  const int id = blockIdx.x;
  const int t  = id % Tq;
  const int bc = id / Tq;
  const int b = bc / NCq, c = bc % NCq;
  const float* cumb = CUM + (size_t)bc * Tq * Sq;
  const int tid = threadIdx.x;
  if (tid < Sq) {
    const float cv = Cc[((size_t)b * LCPq + c * Tq + t) * Sq + tid];
    const float cu = cumb[(size_t)t * Sq + tid];
    sC[tid]  = cv;
    sCt[tid] = cu;
    Ccum[(size_t)bc * Tq * Sq + (size_t)t * Sq + tid] = cv * expf(cu);
  }
  __syncthreads();
  const int tau = tid;
  float acc = 0.0f;
  if (tau <= t) {
    const float* Brow   = Bc + ((size_t)b * LCPq + c * Tq + tau) * Sq;
    const float* cumtau = cumb + (size_t)tau * Sq;
#pragma unroll 8
    for (int s = 0; s < Sq; ++s)
      acc += sC[s] * Brow[s] * expf(sCt[s] - cumtau[s]);
  }
  Mmat[(size_t)bc * Tq * Tq + (size_t)t * Tq + tau] = acc;
}

// ============================================================
// K6: chunk end-state GEMM (WMMA): HLOC[b,c,d,s] = sum_tau xc[tau,d] * wB[tau,s]
//     wB[tau,s] = exp(cum[T-1,s]-cum[tau,s]) * Bc[tau,s]     (staged in LDS)
// grid = B*NC*(D/64); block = 128 (4 waves, each 16 d-rows x 4 s-tiles)
// ============================================================
__global__ __launch_bounds__(128) void k_chunkstate(const float* __restrict__ xc,
                                                    const float* __restrict__ Bc,
                                                    const float* __restrict__ CUM,
                                                    float* __restrict__ HLOC) {
  __shared__ float wB[Tq * Sq];         // 32 KB
  int id = blockIdx.x;
  const int dg = id & 15; id >>= 4;
  const int c = id % NCq, b = id / NCq;
  const int tid = threadIdx.x;
  const float* cum  = CUM + (size_t)(b * NCq + c) * Tq * Sq;
  const float* Brow = Bc + ((size_t)b * LCPq + c * Tq) * Sq;
  for (int i = tid; i < Tq * Sq; i += 128) {
    const int tau = i >> 6, s = i & 63;
    wB[i] = expf(cum[(Tq - 1) * Sq + s] - cum[tau * Sq + s]) * Brow[i];
  }
  __syncthreads();

  const int wave = tid >> 5, lane = tid & 31;
  const int mrow = lane & 15, khalf = (lane >> 4) * 2;
  const int d0 = dg * 64 + wave * 16;
  const float* xb = xc + ((size_t)b * LCPq + c * Tq) * Dq;
  v8f acc[4] = {};
  for (int k0 = 0; k0 < Tq; k0 += 4) {
    const int k = k0 + khalf;
    v2f a;
    a.x = xb[(size_t)k * Dq + d0 + mrow];
    a.y = xb[(size_t)(k + 1) * Dq + d0 + mrow];
#pragma unroll
    for (int nt = 0; nt < 4; ++nt) {
      v2f bf;
      bf.x = wB[k * Sq + nt * 16 + mrow];
      bf.y = wB[(k + 1) * Sq + nt * 16 + mrow];
      acc[nt] = wmma4(a, bf, acc[nt]);
    }
  }
#pragma unroll
  for (int nt = 0; nt < 4; ++nt)
#pragma unroll
    for (int v = 0; v < 8; ++v) {
      const int d = d0 + v + ((lane >> 4) << 3);
      const int s = nt * 16 + mrow;
      HLOC[(((size_t)(b * NCq + c) * Dq) + d) * Sq + s] = acc[nt][v];
    }
}

// ============================================================
// K7: inter-chunk scan (17 steps), in place: HLOC[c] <- H_start[c]
// thread per (b,d,s): carry' = exp(cum_end)*carry + hloc
// ============================================================
__global__ __launch_bounds__(256) void k_interchunk(float* __restrict__ HLOC,
                                                    const float* __restrict__ CUM) {
  const int idx = blockIdx.x * 256 + threadIdx.x;   // B*D*S = 262144
  const int s = idx & 63;
  const int d = (idx >> 6) & 1023;
  const int b = idx >> 16;
  float carry = 0.0f;
  for (int c = 0; c < NCq; ++c) {
    const size_t off = (((size_t)(b * NCq + c) * Dq) + d) * Sq + s;
    const float hl = HLOC[off];
    HLOC[off] = carry;                               // H_start for chunk c
    const float dec = expf(CUM[((size_t)(b * NCq + c) * Tq + (Tq - 1)) * Sq + s]);
    carry = dec * carry + hl;
  }
}

// ============================================================
// K8: output (WMMA): Y = M @ U_chunk + Ccum @ Hstart^T + D_skip .* xc
// grid = B*NC*(D/64); block = 128 (4 waves; wave = 2 t-tiles x 4 d-tiles)
// ============================================================
__global__ __launch_bounds__(128) void k_output(const float* __restrict__ Mmat,
                                                const float* __restrict__ xc,
                                                const float* __restrict__ Ccum,
                                                const float* __restrict__ Hst,
                                                const float* __restrict__ D_skip,
                                                float* __restrict__ out) {
  int id = blockIdx.x;
  const int dg = id & 15; id >>= 4;
  const int c = id % NCq, b = id / NCq;
  const int tid = threadIdx.x;
  const int wave = tid >> 5, lane = tid & 31;
  const int mrow = lane & 15, khalf = (lane >> 4) * 2;

  v8f acc[2][4] = {};
  const float* Mb = Mmat + (size_t)(b * NCq + c) * Tq * Tq;
  const float* xb = xc + ((size_t)b * LCPq + c * Tq) * Dq + dg * 64;

  // GEMM1: [T x T] @ [T x 64]
  for (int k0 = 0; k0 < Tq; k0 += 4) {
    const int k = k0 + khalf;
    v2f bf[4];
#pragma unroll
    for (int nt = 0; nt < 4; ++nt) {
      bf[nt].x = xb[(size_t)k * Dq + nt * 16 + mrow];
      bf[nt].y = xb[(size_t)(k + 1) * Dq + nt * 16 + mrow];
    }
#pragma unroll
    for (int tt = 0; tt < 2; ++tt) {
      const int trow = (wave * 2 + tt) * 16 + mrow;
      v2f a;
      a.x = Mb[(size_t)trow * Tq + k];
      a.y = Mb[(size_t)trow * Tq + k + 1];
#pragma unroll
      for (int nt = 0; nt < 4; ++nt) acc[tt][nt] = wmma4(a, bf[nt], acc[tt][nt]);
    }
  }

  // GEMM2: [T x S] @ [S x 64]  (B operand = Hstart^T)
  const float* Cb = Ccum + (size_t)(b * NCq + c) * Tq * Sq;
  const float* Hb = Hst + ((size_t)(b * NCq + c) * Dq + dg * 64) * Sq;
  for (int k0 = 0; k0 < Sq; k0 += 4) {
    const int k = k0 + khalf;
    v2f bf[4];
#pragma unroll
    for (int nt = 0; nt < 4; ++nt) {
      const float* hp = Hb + (size_t)(nt * 16 + mrow) * Sq + k;
      bf[nt].x = hp[0];
      bf[nt].y = hp[1];
    }
#pragma unroll
    for (int tt = 0; tt < 2; ++tt) {
      const int trow = (wave * 2 + tt) * 16 + mrow;
      v2f a;
      a.x = Cb[(size_t)trow * Sq + k];
      a.y = Cb[(size_t)trow * Sq + k + 1];
#pragma unroll
      for (int nt = 0; nt < 4; ++nt) acc[tt][nt] = wmma4(a, bf[nt], acc[tt][nt]);
    }
  }

  // epilogue: + D_skip[d]*xc, store valid rows only
#pragma unroll
  for (int tt = 0; tt < 2; ++tt)
#pragma unroll
    for (int nt = 0; nt < 4; ++nt)
#pragma unroll
      for (int v = 0; v < 8; ++v) {
        const int t = c * Tq + (wave * 2 + tt) * 16 + v + ((lane >> 4) << 3);
        const int d = dg * 64 + nt * 16 + mrow;
        if (t < LCq) {
          const float u = xc[((size_t)b * LCPq + t) * Dq + d];
          out[((size_t)b * LCq + t) * Dq + d] =
              acc[tt][nt][v] + D_skip[d] * u;
        }
      }
}

// ============================================================
extern "C" void kernel_launch(void* const* d_in, const int* in_sizes, int n_in,
                              void* d_out, int out_size, void* d_ws, size_t ws_size,
                              hipStream_t stream) {
  const float* x      = (const float*)d_in[0];
  const float* a_hat  = (const float*)d_in[1];
  const float* W_U    = (const float*)d_in[2];
  const float* W_B    = (const float*)d_in[3];
  const float* b_B    = (const float*)d_in[4];
  const float* W_C    = (const float*)d_in[5];
  const float* b_C    = (const float*)d_in[6];
  const float* W_d1   = (const float*)d_in[7];
  const float* b_d1   = (const float*)d_in[8];
  const float* W_d2   = (const float*)d_in[9];
  const float* b_d2   = (const float*)d_in[10];
  const float* D_skip = (const float*)d_in[11];
  const float* conv_w = (const float*)d_in[12];
  const float* conv_b = (const float*)d_in[13];
  float* out = (float*)d_out;

  // workspace carve (floats)
  float* ws = (float*)d_ws;
  size_t off = 0;
  float* XC   = ws + off; off += (size_t)Bq * LCPq * Dq;   // 8,912,896
  float* Bc   = ws + off; off += (size_t)Bq * LCPq * Sq;   //   557,056
  float* Cc   = ws + off; off += (size_t)Bq * LCPq * Sq;
  float* LG   = ws + off; off += (size_t)Bq * LCPq * Sq;
  float* CUM  = ws + off; off += (size_t)Bq * NCq * Tq * Sq;
  float* HLOC = ws + off; off += (size_t)Bq * NCq * Dq * Sq; // 4,456,448
  float* MM   = ws + off; off += (size_t)Bq * NCq * Tq * Tq; // 1,114,112
  float* CC2  = ws + off; off += (size_t)Bq * NCq * Tq * Sq;
  (void)off; (void)ws_size; (void)in_sizes; (void)n_in; (void)out_size;

  const int nConv = (Bq * LCPq * Dq + 255) / 256;
  k_conv<<<nConv, 256, 0, stream>>>(x, conv_w, conv_b, XC);

  k_proj<<<(Bq * LCPq) / 16, 128, 0, stream>>>(XC, W_U, W_B, b_B, W_C, b_C, Bc, Cc);

  k_delta<<<(Bq * LCPq) / 8, 256, 0, stream>>>(XC, a_hat, W_d1, b_d1, W_d2, b_d2, LG);

  k_cumsum<<<Bq * NCq, 64, 0, stream>>>(LG, CUM);

  k_segsum<<<Bq * NCq * Tq, 128, 0, stream>>>(Bc, Cc, CUM, MM, CC2);

  k_chunkstate<<<Bq * NCq * (Dq / 64), 128, 0, stream>>>(XC, Bc, CUM, HLOC);

  k_interchunk<<<(Bq * Dq * Sq) / 256, 256, 0, stream>>>(HLOC, CUM);

  k_output<<<Bq * NCq * (Dq / 64), 128, 0, stream>>>(MM, XC, CC2, HLOC, D_skip, out);
}